// DistributionTracker_38113539785054
// MI455X (gfx1250) — compile-verified
//
#include <hip/hip_runtime.h>
#include <stdint.h>

// DistributionTracker forward on MI455X (gfx1250).
// Pure HBM-bandwidth scatter-reduce (1 GiB in -> 1 MB out, ~47us floor at
// 23.3 TB/s). Strategy: LDS-privatized per-class accumulators (250 KB of the
// 320 KB WGP LDS), X tiles staged HBM->LDS with the CDNA5 Tensor Data Mover
// (tensor_load_to_lds, TENSORcnt double-buffer pipeline), bank-conflict-free
// ds_add_f32 accumulation, global_atomic_add_f32 flush.

#define CLASSES     1000
#define DIM         128
#define SLICE_COLS  32
#define NSLICES     4            // DIM / SLICE_COLS
#define TILE_ROWS   128
#define BLOCK       256          // 8 wave32s
#define ROW_CHUNKS  64
#define OUT_COLS    257          // [num | sum(128) | sqr(128)]

typedef __attribute__((ext_vector_type(4))) unsigned int u32x4;
typedef __attribute__((ext_vector_type(8))) int          i32x8;
typedef __attribute__((ext_vector_type(4))) int          i32x4;

// Build a Tensor DMA descriptor (D#) for a 2D tile:
//   tile_dim0 = SLICE_COLS elements (contiguous 128B row segment)
//   tile_dim1 = TILE_ROWS rows, row stride tensor_dim0_stride = DIM elements
//   data_size = 4 bytes, no padding, no multicast, no gather.
// Groups 2/3 (and the trailing descriptor group of the 6-arg builtin) are
// unused for a 2D tensor -> zeros.
__device__ __forceinline__ void tdm_load_tile(const float* gsrc, uint32_t lds_byte_off)
{
    const uint64_t ga = (uint64_t)(uintptr_t)gsrc;

    u32x4 g0;
    g0[0] = 1u;                                            // count=1 (valid), user mode
    g0[1] = lds_byte_off;                                  // LDS dest byte address
    g0[2] = (uint32_t)(ga & 0xFFFFFFFFu);                  // global_addr[31:0]
    g0[3] = (uint32_t)((ga >> 32) & 0x01FFFFFFu)           // global_addr[56:32]
          | (2u << 30);                                    // type = 2 ("image")

    i32x8 g1;
    g1[0] = (int)(2u << 16);                               // data_size = 2 -> 4 bytes
    g1[1] = (int)((uint32_t)SLICE_COLS << 16);             // tensor_dim0[15:0] in [31:16]
    g1[2] = (int)((uint32_t)TILE_ROWS  << 16);             // dim0 hi=0 | tensor_dim1 lo
    g1[3] = (int)((uint32_t)SLICE_COLS << 16);             // dim1 hi=0 | tile_dim0
    g1[4] = (int)TILE_ROWS;                                // tile_dim1 | tile_dim2=0
    g1[5] = (int)DIM;                                      // tensor_dim0_stride[31:0]
    g1[6] = 0;                                             // stride0 hi | dim1_stride lo
    g1[7] = 0;

    const i32x4 z4 = {0, 0, 0, 0};
    const i32x8 z8 = {0, 0, 0, 0, 0, 0, 0, 0};
    __builtin_amdgcn_tensor_load_to_lds(g0, g1, z4, z4, z8, /*cpol=*/0);
}

__global__ void zero_out_kernel(float* __restrict__ out, int n)
{
    int i = blockIdx.x * blockDim.x + threadIdx.x;
    if (i < n) out[i] = 0.0f;
}

__global__ __launch_bounds__(BLOCK, 1)
void dist_tracker_kernel(const float* __restrict__ X,
                         const int*   __restrict__ labels,
                         float*       __restrict__ out,
                         int rows_per_chunk)
{
    // Dynamic LDS layout:
    //   acc : CLASSES * 64 f32 (per-class [sum(32) | sqr(32)])  = 256000 B
    //   cnt : CLASSES f32                                       =   4000 B
    //   lab : TILE_ROWS int                                     =    512 B
    //   xbuf: 2 * TILE_ROWS * SLICE_COLS f32 (double buffer)    =  32768 B
    extern __shared__ float smem[];
    float* acc  = smem;
    float* cnt  = acc + CLASSES * 64;
    int*   lab  = (int*)(cnt + CLASSES);
    float* xbuf = (float*)(lab + TILE_ROWS);

    const int bx   = blockIdx.x;        // row chunk
    const int by   = blockIdx.y;        // column slice
    const int tid  = threadIdx.x;
    const int lane = tid & 31;
    const int rg   = tid >> 5;          // row group 0..7

    for (int i = tid; i < CLASSES * 64; i += BLOCK) acc[i] = 0.0f;
    if (by == 0)
        for (int i = tid; i < CLASSES; i += BLOCK) cnt[i] = 0.0f;

    const int      rowBase = bx * rows_per_chunk;
    const int      T       = rows_per_chunk / TILE_ROWS;
    const float*   xsrc    = X + (size_t)rowBase * DIM + (size_t)by * SLICE_COLS;
    const uint32_t xbufOff = (uint32_t)(uintptr_t)xbuf;   // flat LDS addr low 32 = LDS offset

    // Prologue: wave 0 issues tile 0 (TDM ignores EXEC; executes once per wave).
    if (tid < 32) {
        tdm_load_tile(xsrc, xbufOff);
    }

    for (int t = 0; t < T; ++t) {
        // Stage this tile's labels (previous readers released by end-of-iter barrier).
        if (tid < TILE_ROWS)
            lab[tid] = labels[rowBase + t * TILE_ROWS + tid];

        // Wave 0: prefetch tile t+1 into the other buffer, then wait until
        // tile t is resident (tensor ops complete in order per wave).
        if (tid < 32) {
            if (t + 1 < T) {
                tdm_load_tile(xsrc + (size_t)(t + 1) * TILE_ROWS * DIM,
                              xbufOff + (uint32_t)(((t + 1) & 1) * TILE_ROWS * SLICE_COLS * 4));
                __builtin_amdgcn_s_wait_tensorcnt(1);
            } else {
                __builtin_amdgcn_s_wait_tensorcnt(0);
            }
        }
        __syncthreads();   // tile t data + labels visible to all waves

        const float* xb = xbuf + (t & 1) * TILE_ROWS * SLICE_COLS;
        #pragma unroll 4
        for (int r = rg; r < TILE_ROWS; r += 8) {
            const int   cls = lab[r];
            const float x   = xb[r * SLICE_COLS + lane];
            // lanes 0..31 -> banks 0..31 (sum) and 32..63 (sqr): conflict-free
            atomicAdd(&acc[cls * 64 + lane], x);
            atomicAdd(&acc[cls * 64 + 32 + lane], x * x);
            if (by == 0 && lane == 0)
                atomicAdd(&cnt[cls], 1.0f);
        }
        __syncthreads();   // all reads of tile t done before its buffer is reused
    }

    // Flush per-block partials to global (output is ~1 MB, lives in L2).
    for (int i = tid; i < CLASSES * 64; i += BLOCK) {
        const int cls = i >> 6;
        const int c   = i & 63;
        const int col = (c < 32) ? (1 + by * SLICE_COLS + c)
                                 : (1 + DIM + by * SLICE_COLS + (c - 32));
        atomicAdd(&out[cls * OUT_COLS + col], acc[i]);
    }
    if (by == 0)
        for (int i = tid; i < CLASSES; i += BLOCK)
            atomicAdd(&out[i * OUT_COLS], cnt[i]);
}

extern "C" void kernel_launch(void* const* d_in, const int* in_sizes, int n_in,
                              void* d_out, int out_size, void* d_ws, size_t ws_size,
                              hipStream_t stream)
{
    const float* X      = (const float*)d_in[0];
    const int*   labels = (const int*)d_in[1];
    float*       out    = (float*)d_out;

    const int N = in_sizes[1];                 // number of rows (labels count)
    const int rows_per_chunk = N / ROW_CHUNKS; // 2097152/64 = 32768

    // d_out is poisoned; zero it every call (kernels on one stream serialize).
    zero_out_kernel<<<(out_size + 255) / 256, 256, 0, stream>>>(out, out_size);

    const size_t shmem = (size_t)(CLASSES * 64 + CLASSES) * sizeof(float)
                       + (size_t)TILE_ROWS * sizeof(int)
                       + (size_t)2 * TILE_ROWS * SLICE_COLS * sizeof(float); // 293280 B

    dist_tracker_kernel<<<dim3(ROW_CHUNKS, NSLICES), BLOCK, shmem, stream>>>(
        X, labels, out, rows_per_chunk);
}